// GIN_6055903887407
// MI455X (gfx1250) — compile-verified
//
#include <hip/hip_runtime.h>
#include <hip/hip_bf16.h>
#include <math.h>

// ---------------------------------------------------------------------------
// GIN forward for MI455X (gfx1250, wave32).
//   - All 7 large GEMMs run through V_WMMA_F32_16X16X4_F32 (f32-exact).
//   - LDS images store K-pairs contiguously (v2f) so every WMMA fragment is
//     one aligned ds_load_b64 into an even VGPR pair (no repacking movs).
//   - Edge scatter-add is the roofline bottleneck; it is L2-resident
//     (agg buffer = 25.6 MB << 192 MB L2).
//   - BN (training-mode batch stats) = fused sum/sumsq in GEMM1 epilogue
//     + tiny finalize kernel + fused affine/ReLU in GEMM2 prologue.
// ---------------------------------------------------------------------------

typedef __attribute__((ext_vector_type(2))) float v2f;
typedef __attribute__((ext_vector_type(8))) float v8f;

#define HDIM 128
#define INDIM 100
#define OUTDIM 47
#define TM 64        // rows per block tile
#define KC 32        // K chunk staged in LDS
#define LDA2 18      // As2 row stride in float2 (KC/2 + 2): conflict-free A frags
#define LDB2 144     // Ws2 row stride in float2 (128 + 16): 288 dw = 32 mod 64

// ---------------- prep kernels ----------------

__global__ void pad_x_kernel(const float* __restrict__ x, float* __restrict__ h, int n_rows) {
    int i = blockIdx.x * blockDim.x + threadIdx.x;
    int total = n_rows * HDIM;
    if (i >= total) return;
    int r = i >> 7;
    int c = i & 127;
    h[i] = (c < INDIM) ? x[r * INDIM + c] : 0.0f;
}

__global__ void pad_w_kernel(const float* __restrict__ w, float* __restrict__ wp) {
    int i = blockIdx.x * blockDim.x + threadIdx.x;   // 128*128 elements
    if (i >= HDIM * HDIM) return;
    int k = i >> 7;
    int n = i & 127;
    wp[i] = (k < INDIM) ? w[k * HDIM + n] : 0.0f;
}

__global__ void zero_kernel(float* __restrict__ p, int n) {
    int i = blockIdx.x * blockDim.x + threadIdx.x;
    if (i < n) p[i] = 0.0f;
}

// ---------------- edge scatter: agg[dst] += h[src] ----------------
// One lane per (edge, 4 channels): float4 gather, 4 f32 atomics (L2-resident).

__global__ void scatter_add_kernel(const float* __restrict__ h,
                                   const int* __restrict__ src,
                                   const int* __restrict__ dst,
                                   float* __restrict__ agg, int nE) {
    int t = blockIdx.x * blockDim.x + threadIdx.x;
    int total = nE * 32;
    if (t >= total) return;
    int e = t >> 5;
    int c = (t & 31) << 2;          // channel group: 0,4,...,124
    // lookahead prefetch of a future source row (global_prefetch_b8 path)
    int ep = e + 16;
    if (ep < nE)
        __builtin_prefetch(h + (size_t)src[ep] * HDIM + c, 0, 0);
    int s = src[e], d = dst[e];
    const float4 v = *(const float4*)(h + (size_t)s * HDIM + c);
    float* a = agg + (size_t)d * HDIM + c;
    atomicAdd(a + 0, v.x);
    atomicAdd(a + 1, v.y);
    atomicAdd(a + 2, v.z);
    atomicAdd(a + 3, v.w);
}

// ---------------- WMMA f32 GEMM: out[n_rows][128] = f(A) @ W + bias --------
// Modes:
//   Agg  != null : A_eff = A + Agg                          (GIN aggregate)
//   coef != null : A_eff = relu(A * coef[k] + coef[128+k])  (fused BN+ReLU)
//   stats!= null : accumulate per-channel sum/sumsq of out  (BN batch stats)
//   relu_out     : ReLU on store
// Block: 256 threads = 8 waves; 64 rows x 128 cols; waves in 4x2 grid,
// each wave owns 16 rows x 64 cols = 4 accumulator tiles of 16x16.
// LDS holds K-pairs as v2f: one ds_load_b64 per WMMA fragment.

__global__ __launch_bounds__(256)
void gemm128_wmma_kernel(const float* __restrict__ A,
                         const float* __restrict__ Agg,
                         const float* __restrict__ coef,
                         const float* __restrict__ W,
                         const float* __restrict__ bias,
                         float* __restrict__ out,
                         float* __restrict__ stats,
                         int n_rows, int relu_out) {
    __shared__ v2f As2[TM][LDA2];        // 64*18*8   = 9.2 KB
    __shared__ v2f Ws2[KC / 2][LDB2];    // 16*144*8  = 18.4 KB

    const int tid  = threadIdx.x;
    const int lane = tid & 31;
    const int wid  = tid >> 5;      // 0..7
    const int wr   = wid & 3;       // row tile within block (16 rows)
    const int wc   = wid >> 2;      // column half (64 cols)
    const int row0 = blockIdx.x * TM;
    const int l16  = lane & 15;
    const int hsel = (lane >= 16) ? 1 : 0;   // lo lanes: K-pair j ; hi: j+1

    v8f acc[4];
#pragma unroll
    for (int t = 0; t < 4; ++t) acc[t] = (v8f){0, 0, 0, 0, 0, 0, 0, 0};

    for (int kc = 0; kc < HDIM; kc += KC) {
        // ---- stage A tile: 64 rows x KC cols; float4 load -> two v2f pairs
        for (int i = tid; i < TM * (KC / 4); i += 256) {
            int r  = i / (KC / 4);
            int c4 = (i % (KC / 4)) * 4;           // k offset within chunk
            int gr = row0 + r;
            float4 v = make_float4(0.f, 0.f, 0.f, 0.f);
            if (gr < n_rows) {
                v = *(const float4*)(A + (size_t)gr * HDIM + kc + c4);
                if (Agg) {
                    float4 w = *(const float4*)(Agg + (size_t)gr * HDIM + kc + c4);
                    v.x += w.x; v.y += w.y; v.z += w.z; v.w += w.w;
                }
                if (coef) {
                    int k = kc + c4;
                    v.x = fmaxf(v.x * coef[k + 0] + coef[HDIM + k + 0], 0.f);
                    v.y = fmaxf(v.y * coef[k + 1] + coef[HDIM + k + 1], 0.f);
                    v.z = fmaxf(v.z * coef[k + 2] + coef[HDIM + k + 2], 0.f);
                    v.w = fmaxf(v.w * coef[k + 3] + coef[HDIM + k + 3], 0.f);
                }
            }
            As2[r][c4 / 2 + 0] = (v2f){v.x, v.y};
            As2[r][c4 / 2 + 1] = (v2f){v.z, v.w};
        }
        // ---- stage W slab: KC rows x 128 cols, K-paired:
        //      Ws2[j][n] = { W[kc+2j][n], W[kc+2j+1][n] }
        for (int i = tid; i < (KC / 2) * (HDIM / 4); i += 256) {
            int j  = i / (HDIM / 4);               // K-pair index
            int c4 = (i % (HDIM / 4)) * 4;
            float4 r0 = *(const float4*)(W + (size_t)(kc + 2 * j + 0) * HDIM + c4);
            float4 r1 = *(const float4*)(W + (size_t)(kc + 2 * j + 1) * HDIM + c4);
            Ws2[j][c4 + 0] = (v2f){r0.x, r1.x};
            Ws2[j][c4 + 1] = (v2f){r0.y, r1.y};
            Ws2[j][c4 + 2] = (v2f){r0.z, r1.z};
            Ws2[j][c4 + 3] = (v2f){r0.w, r1.w};
        }
        __syncthreads();

        // ---- WMMA inner loop: K in steps of 4 (= 2 K-pairs)
        const int ar0 = wr * 16;
#pragma unroll
        for (int j = 0; j < KC / 2; j += 2) {
            // A fragment (ISA 32-bit A 16x4): lanes 0-15 K-pair j, 16-31 pair j+1
            v2f a = As2[ar0 + l16][j + hsel];
#pragma unroll
            for (int t = 0; t < 4; ++t) {
                int c0 = wc * 64 + t * 16;
                v2f b = Ws2[j + hsel][c0 + l16];
                acc[t] = __builtin_amdgcn_wmma_f32_16x16x4_f32(
                    false, a, false, b, (short)0, acc[t], false, false);
            }
        }
        __syncthreads();
    }

    // ---- epilogue: bias, optional ReLU, store + optional BN-stat accumulate
    // C/D 16x16 layout: N = lane%16 ; M = vgpr + (lane>=16 ? 8 : 0)
    const int mofs = hsel * 8;
#pragma unroll
    for (int t = 0; t < 4; ++t) {
        int col = wc * 64 + t * 16 + l16;
        float bv = bias ? bias[col] : 0.f;
        float s = 0.f, s2 = 0.f;
#pragma unroll
        for (int v = 0; v < 8; ++v) {
            int gr = row0 + wr * 16 + mofs + v;
            if (gr < n_rows) {
                float o = acc[t][v] + bv;
                if (relu_out) o = fmaxf(o, 0.f);
                out[(size_t)gr * HDIM + col] = o;
                s += o; s2 += o * o;
            }
        }
        if (stats) {
            atomicAdd(&stats[col], s);
            atomicAdd(&stats[HDIM + col], s2);
        }
    }
}

// ---------------- BN finalize: fold mu/var/gamma/beta into affine ----------

__global__ void bn_finalize_kernel(const float* __restrict__ stats,
                                   const float* __restrict__ g,
                                   const float* __restrict__ be,
                                   float* __restrict__ coef, int n_rows) {
    int c = threadIdx.x;
    if (c < HDIM) {
        float inv_n = 1.0f / (float)n_rows;
        float mu  = stats[c] * inv_n;
        float var = stats[HDIM + c] * inv_n - mu * mu;
        float a   = g[c] * rsqrtf(var + 1e-5f);
        coef[c]        = a;
        coef[HDIM + c] = be[c] - mu * a;
    }
}

// ---------------- head: lin2 (128x47) + log_softmax, one wave per row ------

__global__ void head_logsoftmax_kernel(const float* __restrict__ h,
                                       const float* __restrict__ W,
                                       const float* __restrict__ b,
                                       float* __restrict__ out, int n_rows) {
    int row  = blockIdx.x * (blockDim.x >> 5) + (threadIdx.x >> 5);
    int lane = threadIdx.x & 31;
    if (row >= n_rows) return;
    const float* hr = h + (size_t)row * HDIM;
    const bool two = (lane + 32) < OUTDIM;
    float acc0 = b[lane];
    float acc1 = two ? b[lane + 32] : 0.f;
    for (int k = 0; k < HDIM; ++k) {
        float hv = hr[k];
        acc0 += hv * W[k * OUTDIM + lane];
        if (two) acc1 += hv * W[k * OUTDIM + lane + 32];
    }
    float m = two ? fmaxf(acc0, acc1) : acc0;
#pragma unroll
    for (int o = 16; o > 0; o >>= 1) m = fmaxf(m, __shfl_xor(m, o, 32));
    float e = __expf(acc0 - m) + (two ? __expf(acc1 - m) : 0.f);
#pragma unroll
    for (int o = 16; o > 0; o >>= 1) e += __shfl_xor(e, o, 32);
    float lse = __logf(e) + m;
    out[(size_t)row * OUTDIM + lane] = acc0 - lse;
    if (two) out[(size_t)row * OUTDIM + lane + 32] = acc1 - lse;
}

// ---------------- orchestration ----------------

extern "C" void kernel_launch(void* const* d_in, const int* in_sizes, int n_in,
                              void* d_out, int out_size, void* d_ws, size_t ws_size,
                              hipStream_t stream) {
    const float* x   = (const float*)d_in[0];
    const int*   src = (const int*)d_in[1];
    const int*   dst = (const int*)d_in[2];
    const int N = in_sizes[0] / INDIM;
    const int E = in_sizes[1];

    const float *W1[3], *b1[3], *g[3], *be[3], *W2[3], *b2[3];
    for (int l = 0; l < 3; ++l) {
        int base = 3 + l * 6;
        W1[l] = (const float*)d_in[base + 0];
        b1[l] = (const float*)d_in[base + 1];
        g[l]  = (const float*)d_in[base + 2];
        be[l] = (const float*)d_in[base + 3];
        W2[l] = (const float*)d_in[base + 4];
        b2[l] = (const float*)d_in[base + 5];
    }
    const float* lin1_W = (const float*)d_in[21];
    const float* lin1_b = (const float*)d_in[22];
    const float* lin2_W = (const float*)d_in[23];
    const float* lin2_b = (const float*)d_in[24];

    // workspace layout (floats): hA | hB | agg | stats(256) | coef(256) | W1_0p
    float* ws    = (float*)d_ws;
    float* hA    = ws;
    float* hB    = hA + (size_t)N * HDIM;
    float* agg   = hB + (size_t)N * HDIM;
    float* stats = agg + (size_t)N * HDIM;     // contiguous with agg for one zero pass
    float* coef  = stats + 2 * HDIM;
    float* W10p  = coef + 2 * HDIM;

    const int threads = 256;
    const int nH      = N * HDIM;
    dim3 blk(threads);

    // prep: pad x -> hA, pad W1_0 -> 128x128
    pad_x_kernel<<<dim3((nH + threads - 1) / threads), blk, 0, stream>>>(x, hA, N);
    pad_w_kernel<<<dim3((HDIM * HDIM + threads - 1) / threads), blk, 0, stream>>>(W1[0], W10p);

    const int gemm_grid    = (N + TM - 1) / TM;
    const int scatter_grid = (E * 32 + threads - 1) / threads;

    for (int l = 0; l < 3; ++l) {
        // zero agg + stats (contiguous)
        zero_kernel<<<dim3((nH + 2 * HDIM + threads - 1) / threads), blk, 0, stream>>>(
            agg, nH + 2 * HDIM);
        // agg[dst] += h[src]
        scatter_add_kernel<<<dim3(scatter_grid), blk, 0, stream>>>(hA, src, dst, agg, E);
        // hB = (hA + agg) @ W1 + b1 ; accumulate BN stats
        gemm128_wmma_kernel<<<dim3(gemm_grid), blk, 0, stream>>>(
            hA, agg, nullptr, (l == 0 ? W10p : W1[l]), b1[l], hB, stats, N, 0);
        // fold BN into per-channel affine
        bn_finalize_kernel<<<dim3(1), dim3(HDIM), 0, stream>>>(stats, g[l], be[l], coef, N);
        // hA = relu( relu(BN(hB)) @ W2 + b2 )
        gemm128_wmma_kernel<<<dim3(gemm_grid), blk, 0, stream>>>(
            hB, nullptr, coef, W2[l], b2[l], hA, nullptr, N, 1);
    }

    // hB = relu(hA @ lin1_W + lin1_b)
    gemm128_wmma_kernel<<<dim3(gemm_grid), blk, 0, stream>>>(
        hA, nullptr, nullptr, lin1_W, lin1_b, hB, nullptr, N, 1);

    // out = log_softmax(hB @ lin2_W + lin2_b)
    const int rows_per_blk = threads / 32;
    head_logsoftmax_kernel<<<dim3((N + rows_per_blk - 1) / rows_per_blk), blk, 0, stream>>>(
        hB, lin2_W, lin2_b, (float*)d_out, N);
}